// GCNProbe_83339545411793
// MI455X (gfx1250) — compile-verified
//
#include <hip/hip_runtime.h>

typedef __attribute__((ext_vector_type(2))) float v2f;
typedef __attribute__((ext_vector_type(8))) float v8f;

#define HID 64
#define NGRAPH 64

// ---------------------------------------------------------------------------
// GEMM: out[r, 0:64] = A[xmap[r], 0:64] @ W[64,64]   (fp32 WMMA 16x16x4)
// One wave per 16-row stripe; 4 column tiles x 16 K-steps of v_wmma.
// A-frag layout (16x4 f32): lanes 0-15 -> K=k0+{0,1}; lanes 16-31 -> K=k0+{2,3}
// B-frag layout (4x16 f32): VGPR0 lanes0-15 = row K=k0 (lanes16-31 K=k0+2),
//                           VGPR1 = K=k0+1 (K=k0+3)
// D layout (16x16 f32):  c[v] holds M = v + 8*(lane>=16), N = lane&15
// ---------------------------------------------------------------------------
__global__ __launch_bounds__(256) void gemm_wmma_f32(
    const float* __restrict__ A, const float* __restrict__ W,
    float* __restrict__ out, const int* __restrict__ xmap, int nrows) {
  const int lane    = threadIdx.x & 31;
  const int waveId  = threadIdx.x >> 5;
  const int tileM   = blockIdx.x * 8 + waveId;      // 16-row tile index
  if (tileM * 16 >= nrows) return;                  // wave-uniform: EXEC stays all-1s

  const int half = lane >> 4;                       // 0 or 1
  const int l    = lane & 15;
  const int row  = tileM * 16 + l;
  const int arow = xmap ? xmap[row] : row;
  const float* __restrict__ Arow = A + (long)arow * HID;

  // Preload all 16 A fragments (covers K = 0..63 for this lane's rows)
  v2f afrag[16];
#pragma unroll
  for (int kk = 0; kk < 16; ++kk) {
    const int kb = kk * 4 + half * 2;
    afrag[kk].x = Arow[kb];
    afrag[kk].y = Arow[kb + 1];
  }

#pragma unroll
  for (int nt = 0; nt < 4; ++nt) {
    const int n0 = nt * 16;
    v8f c = {0.f, 0.f, 0.f, 0.f, 0.f, 0.f, 0.f, 0.f};
#pragma unroll
    for (int kk = 0; kk < 16; ++kk) {
      const int krow = kk * 4 + half * 2;
      v2f b;
      b.x = W[krow * HID + n0 + l];
      b.y = W[(krow + 1) * HID + n0 + l];
      c = __builtin_amdgcn_wmma_f32_16x16x4_f32(
          /*neg_a=*/false, afrag[kk], /*neg_b=*/false, b,
          /*c_mod=*/(short)0, c, /*reuse_a=*/false, /*reuse_b=*/false);
    }
#pragma unroll
    for (int v = 0; v < 8; ++v) {
      const int m = tileM * 16 + v + half * 8;
      out[(long)m * HID + n0 + l] = c[v];
    }
  }
}

// ---------------------------------------------------------------------------
// Zero fill
// ---------------------------------------------------------------------------
__global__ __launch_bounds__(256) void zero_f32(float* __restrict__ p, long n) {
  long t = (long)blockIdx.x * blockDim.x + threadIdx.x;
  if (t < n) p[t] = 0.0f;
}

// ---------------------------------------------------------------------------
// Edge scatter-add: h[dst[e], c] += m[src[e], c] * w[e]
// One thread = 2 channels of one edge; a wave covers one edge's full 64
// channels -> 256B contiguous gather + contiguous f32 atomics (resolve in L2).
// ---------------------------------------------------------------------------
__global__ __launch_bounds__(256) void scatter_edges(
    const float* __restrict__ m, const int* __restrict__ src,
    const int* __restrict__ dst, const float* __restrict__ ew,
    float* __restrict__ hout, long total) {
  long t = (long)blockIdx.x * blockDim.x + threadIdx.x;
  if (t >= total) return;
  const int e  = (int)(t >> 5);
  const int c2 = (int)(t & 31) * 2;
  const float w = ew[e];
  const int s = src[e];
  const int d = dst[e];
  const float2 v = *(const float2*)(m + (long)s * HID + c2);
  atomicAdd(&hout[(long)d * HID + c2],     v.x * w);
  atomicAdd(&hout[(long)d * HID + c2 + 1], v.y * w);
}

// ---------------------------------------------------------------------------
// Bias + LayerNorm + ReLU, in place. One wave32 per node, 2 channels/lane.
// ---------------------------------------------------------------------------
__global__ __launch_bounds__(256) void bias_ln_relu(
    float* __restrict__ h, const float* __restrict__ bias,
    const float* __restrict__ g, const float* __restrict__ b, int nnodes) {
  const int node = blockIdx.x * (blockDim.x >> 5) + (threadIdx.x >> 5);
  const int lane = threadIdx.x & 31;
  if (node >= nnodes) return;
  float2* hp = (float2*)(h + (long)node * HID);
  float2 v = hp[lane];
  v.x += bias[lane * 2];
  v.y += bias[lane * 2 + 1];
  float s = v.x + v.y;
#pragma unroll
  for (int mk = 16; mk >= 1; mk >>= 1) s += __shfl_xor(s, mk, 32);
  const float mu = s * (1.0f / 64.0f);
  const float dx = v.x - mu, dy = v.y - mu;
  float q = dx * dx + dy * dy;
#pragma unroll
  for (int mk = 16; mk >= 1; mk >>= 1) q += __shfl_xor(q, mk, 32);
  const float rstd = rsqrtf(q * (1.0f / 64.0f) + 1e-5f);
  float2 o;
  o.x = fmaxf(dx * rstd * g[lane * 2]     + b[lane * 2],     0.0f);
  o.y = fmaxf(dy * rstd * g[lane * 2 + 1] + b[lane * 2 + 1], 0.0f);
  hp[lane] = o;
}

// ---------------------------------------------------------------------------
// Per-graph pooling accumulation. h >= 0 (post-ReLU) so int-bit atomicMax is
// order-preserving and a 0-initialized max buffer matches the reference's
// where(counts>0, max, 0).
// ---------------------------------------------------------------------------
__global__ __launch_bounds__(256) void pool_acc(
    const float* __restrict__ h, const int* __restrict__ batch,
    float* __restrict__ sum, float* __restrict__ mx,
    float* __restrict__ counts, int nnodes) {
  long t = (long)blockIdx.x * blockDim.x + threadIdx.x;
  if (t >= (long)nnodes * 32) return;
  const int n  = (int)(t >> 5);
  const int c2 = (int)(t & 31) * 2;
  const int gph = batch[n];
  const float2 v = *(const float2*)(h + (long)n * HID + c2);
  atomicAdd(&sum[gph * HID + c2],     v.x);
  atomicAdd(&sum[gph * HID + c2 + 1], v.y);
  atomicMax((int*)&mx[gph * HID + c2],     __float_as_int(v.x));
  atomicMax((int*)&mx[gph * HID + c2 + 1], __float_as_int(v.y));
  if (c2 == 0) atomicAdd(&counts[gph], 1.0f);
}

// ---------------------------------------------------------------------------
// Readout MLP: g=[mean|max] (128) -> relu(g@W1+b1) (64) -> @W2+b2 -> out[64]
// One 64-thread block per graph (tiny).
// ---------------------------------------------------------------------------
__global__ __launch_bounds__(64) void mlp_readout(
    const float* __restrict__ sum, const float* __restrict__ mx,
    const float* __restrict__ counts, const float* __restrict__ W1,
    const float* __restrict__ b1, const float* __restrict__ W2,
    const float* __restrict__ b2, float* __restrict__ out) {
  const int gph = blockIdx.x;
  const int j = threadIdx.x;
  __shared__ float gfeat[2 * HID];
  __shared__ float hid[HID];
  const float cnt = fmaxf(counts[gph], 1.0f);
  gfeat[j]        = sum[gph * HID + j] / cnt;
  gfeat[HID + j]  = mx[gph * HID + j];
  __syncthreads();
  float acc = b1[j];
#pragma unroll 8
  for (int k = 0; k < 2 * HID; ++k) acc += gfeat[k] * W1[k * HID + j];
  hid[j] = fmaxf(acc, 0.0f) * W2[j];
  __syncthreads();
  if (j < 32) hid[j] += hid[j + 32];
  __syncthreads();
  if (j == 0) {
    float r = 0.0f;
#pragma unroll
    for (int k = 0; k < 32; ++k) r += hid[k];
    out[gph] = r + b2[0];
  }
}

// ---------------------------------------------------------------------------
extern "C" void kernel_launch(void* const* d_in, const int* in_sizes, int n_in,
                              void* d_out, int out_size, void* d_ws, size_t ws_size,
                              hipStream_t stream) {
  const int*   x     = (const int*)d_in[0];
  const int*   eidx  = (const int*)d_in[1];
  const float* ew    = (const float*)d_in[2];
  const int*   batch = (const int*)d_in[3];
  const float* emb   = (const float*)d_in[4];
  const float* convW = (const float*)d_in[5];
  const float* convB = (const float*)d_in[6];
  const float* lnG   = (const float*)d_in[7];
  const float* lnB   = (const float*)d_in[8];
  const float* W1    = (const float*)d_in[9];
  const float* b1    = (const float*)d_in[10];
  const float* W2    = (const float*)d_in[11];
  const float* b2    = (const float*)d_in[12];

  const int NN = in_sizes[0];          // 50000 nodes
  const int NE = in_sizes[1] / 2;      // 1.25M edges
  const int* src = eidx;
  const int* dst = eidx + NE;

  // Workspace layout (floats)
  float* ws   = (float*)d_ws;
  float* bufM = ws;                               // [NN,64]  GEMM output m
  float* bufH = bufM + (long)NN * HID;            // [NN,64]  aggregated h
  float* psum = bufH + (long)NN * HID;            // [64,64]
  float* pmax = psum + NGRAPH * HID;              // [64,64]
  float* pcnt = pmax + NGRAPH * HID;              // [64]

  const int tilesM     = NN / 16 + (NN % 16 ? 1 : 0);
  const int gemmBlocks = (tilesM + 7) / 8;
  const long hElems    = (long)NN * HID;
  const int  zeroBlk   = (int)((hElems + 255) / 256);
  const long scatTot   = (long)NE * 32;
  const int  scatBlk   = (int)((scatTot + 255) / 256);
  const int  lnBlk     = (NN + 7) / 8;
  const long poolTot   = (long)NN * 32;
  const int  poolBlk   = (int)((poolTot + 255) / 256);

  const float* hin  = emb;
  const int*   xmap = x;     // layer 0: h = emb[x]; x is identity but honor it
  for (int L = 0; L < 3; ++L) {
    gemm_wmma_f32<<<gemmBlocks, 256, 0, stream>>>(hin, convW + L * HID * HID,
                                                  bufM, xmap, NN);
    zero_f32<<<zeroBlk, 256, 0, stream>>>(bufH, hElems);
    scatter_edges<<<scatBlk, 256, 0, stream>>>(bufM, src, dst, ew, bufH, scatTot);
    bias_ln_relu<<<lnBlk, 256, 0, stream>>>(bufH, convB + L * HID,
                                            lnG + L * HID, lnB + L * HID, NN);
    hin = bufH;
    xmap = nullptr;
  }

  // Pooling buffers: zero-init (0.0f is also correct for max since h >= 0)
  zero_f32<<<(2 * NGRAPH * HID + NGRAPH + 255) / 256, 256, 0, stream>>>(
      psum, 2L * NGRAPH * HID + NGRAPH);
  pool_acc<<<poolBlk, 256, 0, stream>>>(bufH, batch, psum, pmax, pcnt, NN);

  mlp_readout<<<NGRAPH, 64, 0, stream>>>(psum, pmax, pcnt, W1, b1, W2, b2,
                                         (float*)d_out);
}